// BlueGIN_79474074845433
// MI455X (gfx1250) — compile-verified
//
#include <hip/hip_runtime.h>
#include <hip/hip_bf16.h>

typedef __attribute__((ext_vector_type(2))) float f32x2;
typedef __attribute__((ext_vector_type(8))) float f32x8;

#define N_NODES 200000
#define N_EDGES 1000000
#define NUM_GRAPHS 256
#define NODE_DIM 10
#define HID 64

// ---------------------------------------------------------------------------
// Elementwise init kernels
// ---------------------------------------------------------------------------
__global__ void init_z1_kernel(const float* __restrict__ x,
                               const float* __restrict__ eps1,
                               float* __restrict__ z1) {
    long idx = (long)blockIdx.x * blockDim.x + threadIdx.x;
    long total = (long)N_NODES * 12;
    if (idx >= total) return;
    long node = idx / 12;
    int  col  = (int)(idx - node * 12);
    float v = 0.0f;
    if (col < NODE_DIM) v = (1.0f + eps1[0]) * x[node * NODE_DIM + col];
    z1[idx] = v;
}

__global__ void init_z2_kernel(const float* __restrict__ h1,
                               const float* __restrict__ eps2,
                               float* __restrict__ z2) {
    long idx = (long)blockIdx.x * blockDim.x + threadIdx.x;  // float4 index
    long total = ((long)N_NODES * HID) / 4;
    if (idx >= total) return;
    float4 v = *(const float4*)(h1 + idx * 4);
    float s = 1.0f + eps2[0];
    v.x *= s; v.y *= s; v.z *= s; v.w *= s;
    *(float4*)(z2 + idx * 4) = v;
}

__global__ void zero_pool_kernel(float* __restrict__ pool) {
    int idx = blockIdx.x * blockDim.x + threadIdx.x;
    if (idx < NUM_GRAPHS * HID) pool[idx] = 0.0f;
}

// ---------------------------------------------------------------------------
// Edge scatter-add kernels (global_atomic_add_f32, L2-resident destination)
// ---------------------------------------------------------------------------
__global__ void edge_agg1_kernel(const int* __restrict__ src,
                                 const int* __restrict__ dst,
                                 const float* __restrict__ x,
                                 float* __restrict__ z1) {
    long idx = (long)blockIdx.x * blockDim.x + threadIdx.x;
    long total = (long)N_EDGES * NODE_DIM;
    if (idx >= total) return;
    long e = idx / NODE_DIM;
    int  c = (int)(idx - e * NODE_DIM);
    int s = src[e];
    int d = dst[e];
    atomicAdd(&z1[(long)d * 12 + c], x[(long)s * NODE_DIM + c]);
}

__global__ void edge_agg2_kernel(const int* __restrict__ src,
                                 const int* __restrict__ dst,
                                 const float* __restrict__ h1,
                                 float* __restrict__ z2) {
    long idx = (long)blockIdx.x * blockDim.x + threadIdx.x;  // (edge, col-group)
    long total = (long)N_EDGES * (HID / 4);
    if (idx >= total) return;
    long e = idx >> 4;
    int  q = (int)(idx & 15);
    int s = src[e];
    int d = dst[e];
    float4 v = *(const float4*)(h1 + (long)s * HID + q * 4);
    float* zp = z2 + (long)d * HID + q * 4;
    atomicAdd(zp + 0, v.x);
    atomicAdd(zp + 1, v.y);
    atomicAdd(zp + 2, v.z);
    atomicAdd(zp + 3, v.w);
}

// ---------------------------------------------------------------------------
// Fused GIN MLP: out = relu( relu(z @ Wa + ba) @ Wb + bb )
// One wave handles a 16-node tile; WMMA f32 16x16x4 throughout.
//   KSTEPS : K/4 steps of the first GEMM (3 for padded K=12, 16 for K=64)
//   IN_ROWS: valid rows of Wa (10 or 64); rest zero-padded in LDS
//   DO_POOL: true -> atomic global-add-pool by batch id, false -> write h1
// ---------------------------------------------------------------------------
template <int KSTEPS, int IN_ROWS, bool DO_POOL>
__global__ __launch_bounds__(128)
void gin_mlp_kernel(const float* __restrict__ zin,
                    const float* __restrict__ Wa, const float* __restrict__ ba,
                    const float* __restrict__ Wb, const float* __restrict__ bb,
                    float* __restrict__ outp,
                    const int* __restrict__ batch) {
    __shared__ float sWa[KSTEPS * 4 * 64];
    __shared__ float sBa[64];
    __shared__ float sWb[64 * 64];
    __shared__ float sBb[64];
    __shared__ float sT[4][16 * 64];   // per-wave tile staging

    const int tid = threadIdx.x;

    // ---- block-wide weight staging into LDS ----
    for (int i = tid; i < KSTEPS * 4 * 64; i += 128)
        sWa[i] = (i < IN_ROWS * 64) ? Wa[i] : 0.0f;
    for (int i = tid; i < 64 * 64; i += 128)
        sWb[i] = Wb[i];
    if (tid < 64) { sBa[tid] = ba[tid]; sBb[tid] = bb[tid]; }
    __syncthreads();

    const int wave = tid >> 5;
    const int lane = tid & 31;
    const int half = lane >> 4;      // 0: lanes 0-15 (K 0,1) ; 1: lanes 16-31 (K 2,3)
    const int m    = lane & 15;      // matrix row within tile
    const long tile = (long)blockIdx.x * 4 + wave;   // 12500 tiles exactly
    const long base = tile * 16;

    float* tls = &sT[wave][0];

    // ---- GEMM 1: t = relu(z @ Wa + ba), A fragments straight from global ----
    f32x2 aF[KSTEPS];
    {
        const float* rowp = zin + (base + m) * (KSTEPS * 4) + 2 * half;
#pragma unroll
        for (int k = 0; k < KSTEPS; ++k)
            aF[k] = *(const f32x2*)(rowp + 4 * k);
    }

#pragma unroll
    for (int n = 0; n < 4; ++n) {
        const float bias = sBa[n * 16 + m];
        f32x8 c;
#pragma unroll
        for (int j = 0; j < 8; ++j) c[j] = bias;
#pragma unroll
        for (int k = 0; k < KSTEPS; ++k) {
            const int row0 = 4 * k + 2 * half;
            f32x2 b;
            b.x = sWa[row0 * 64 + n * 16 + m];
            b.y = sWa[row0 * 64 + 64 + n * 16 + m];
            c = __builtin_amdgcn_wmma_f32_16x16x4_f32(
                    false, aF[k], false, b, (short)0, c, false, false);
        }
#pragma unroll
        for (int j = 0; j < 8; ++j) {
            float v = fmaxf(c[j], 0.0f);
            tls[(j + 8 * half) * 64 + n * 16 + m] = v;   // C/D layout -> row major
        }
    }

    // ---- GEMM 2: out = relu(t @ Wb + bb); A fragments re-read from LDS ----
    f32x2 a2[16];
#pragma unroll
    for (int k = 0; k < 16; ++k)
        a2[k] = *(const f32x2*)&tls[m * 64 + 4 * k + 2 * half];

#pragma unroll
    for (int n = 0; n < 4; ++n) {
        const float bias = sBb[n * 16 + m];
        f32x8 c;
#pragma unroll
        for (int j = 0; j < 8; ++j) c[j] = bias;
#pragma unroll
        for (int k = 0; k < 16; ++k) {
            const int row0 = 4 * k + 2 * half;
            f32x2 b;
            b.x = sWb[row0 * 64 + n * 16 + m];
            b.y = sWb[row0 * 64 + 64 + n * 16 + m];
            c = __builtin_amdgcn_wmma_f32_16x16x4_f32(
                    false, a2[k], false, b, (short)0, c, false, false);
        }
#pragma unroll
        for (int j = 0; j < 8; ++j) {
            float v = fmaxf(c[j], 0.0f);
            tls[(j + 8 * half) * 64 + n * 16 + m] = v;
        }
    }

    // ---- emit: either coalesced h1 write or fused global-add-pool ----
    if (DO_POOL) {
        for (int i = lane; i < 16 * 64; i += 32) {
            int node = i >> 6;
            int col  = i & 63;
            int g = batch[base + node];
            atomicAdd(&outp[g * HID + col], tls[i]);
        }
    } else {
        float* dstp = outp + base * HID;
        for (int i = lane; i < 256; i += 32)
            *(float4*)(dstp + i * 4) = *(const float4*)&tls[i * 4];
    }
}

// ---------------------------------------------------------------------------
// Heads: presence = sigmoid(relu(pool@Wp1+bp1)@Wp2+bp2), order = pool@Wo+bo
// One block per graph, 64 threads.
// ---------------------------------------------------------------------------
__global__ __launch_bounds__(64)
void heads_kernel(const float* __restrict__ pool,
                  const float* __restrict__ Wp1, const float* __restrict__ bp1,
                  const float* __restrict__ Wp2, const float* __restrict__ bp2,
                  const float* __restrict__ Wo,  const float* __restrict__ bo,
                  float* __restrict__ out) {
    __shared__ float sp[64];
    __shared__ float st[32];
    const int g = blockIdx.x;
    const int tid = threadIdx.x;

    sp[tid] = pool[g * HID + tid];
    __syncthreads();

    if (tid < 32) {
        float s = bp1[tid];
        for (int k = 0; k < 64; ++k) s += sp[k] * Wp1[k * 32 + tid];
        st[tid] = fmaxf(s, 0.0f);
    }
    __syncthreads();

    if (tid < 8) {
        float s = bp2[tid];
        for (int k = 0; k < 32; ++k) s += st[k] * Wp2[k * 8 + tid];
        out[g * 8 + tid] = 1.0f / (1.0f + expf(-s));          // presence
    }
    if (tid < 24) {
        float s = bo[tid];
        for (int k = 0; k < 64; ++k) s += sp[k] * Wo[k * 24 + tid];
        out[NUM_GRAPHS * 8 + g * 24 + tid] = s;               // order
    }
}

// ---------------------------------------------------------------------------
// Launch
// ---------------------------------------------------------------------------
extern "C" void kernel_launch(void* const* d_in, const int* in_sizes, int n_in,
                              void* d_out, int out_size, void* d_ws, size_t ws_size,
                              hipStream_t stream) {
    const float* x    = (const float*)d_in[0];
    const int*   ei   = (const int*)d_in[1];
    const int*   bat  = (const int*)d_in[2];
    const float* W1a  = (const float*)d_in[3];
    const float* b1a  = (const float*)d_in[4];
    const float* W1b  = (const float*)d_in[5];
    const float* b1b  = (const float*)d_in[6];
    const float* eps1 = (const float*)d_in[7];
    const float* W2a  = (const float*)d_in[8];
    const float* b2a  = (const float*)d_in[9];
    const float* W2b  = (const float*)d_in[10];
    const float* b2b  = (const float*)d_in[11];
    const float* eps2 = (const float*)d_in[12];
    const float* Wp1  = (const float*)d_in[13];
    const float* bp1  = (const float*)d_in[14];
    const float* Wp2  = (const float*)d_in[15];
    const float* bp2  = (const float*)d_in[16];
    const float* Wo   = (const float*)d_in[17];
    const float* bo   = (const float*)d_in[18];
    float* out = (float*)d_out;

    const int* src = ei;            // edge_index[0]
    const int* dst = ei + N_EDGES;  // edge_index[1]

    // workspace layout (floats)
    float* ws   = (float*)d_ws;
    float* z1   = ws;                                   // 200000 * 12
    float* h1   = z1 + (long)N_NODES * 12;              // 200000 * 64
    float* z2   = h1 + (long)N_NODES * HID;             // 200000 * 64
    float* pool = z2 + (long)N_NODES * HID;             // 256 * 64

    // ---- layer 1 ----
    {
        long total = (long)N_NODES * 12;
        init_z1_kernel<<<(int)((total + 255) / 256), 256, 0, stream>>>(x, eps1, z1);
    }
    {
        long total = (long)N_EDGES * NODE_DIM;
        edge_agg1_kernel<<<(int)((total + 255) / 256), 256, 0, stream>>>(src, dst, x, z1);
    }
    gin_mlp_kernel<3, NODE_DIM, false><<<N_NODES / 64, 128, 0, stream>>>(
        z1, W1a, b1a, W1b, b1b, h1, nullptr);

    // ---- layer 2 + fused pooling ----
    {
        long total = ((long)N_NODES * HID) / 4;
        init_z2_kernel<<<(int)((total + 255) / 256), 256, 0, stream>>>(h1, eps2, z2);
    }
    {
        long total = (long)N_EDGES * (HID / 4);
        edge_agg2_kernel<<<(int)((total + 255) / 256), 256, 0, stream>>>(src, dst, h1, z2);
    }
    zero_pool_kernel<<<(NUM_GRAPHS * HID + 255) / 256, 256, 0, stream>>>(pool);
    gin_mlp_kernel<16, HID, true><<<N_NODES / 64, 128, 0, stream>>>(
        z2, W2a, b2a, W2b, b2b, pool, bat);

    // ---- heads ----
    heads_kernel<<<NUM_GRAPHS, 64, 0, stream>>>(pool, Wp1, bp1, Wp2, bp2, Wo, bo, out);
}